// CrossAttentionFusion_6038724018592
// MI455X (gfx1250) — compile-verified
//
#include <hip/hip_runtime.h>
#include <math.h>

typedef __attribute__((ext_vector_type(2))) float v2f;
typedef __attribute__((ext_vector_type(8))) float v8f;
typedef __attribute__((ext_vector_type(4))) unsigned int u32x4;
typedef __attribute__((ext_vector_type(8))) int i32x8;
typedef __attribute__((ext_vector_type(4))) int i32x4;

#define NH  16
#define HD  128
#define B_  8
#define LQ  1024
#define LK  576
#define QD  2048
#define KD  1024
#define H_  2048

// ---------------------------------------------------------------------------
// Issue a 2D TDM tile load (global -> LDS) via tensor_load_to_lds.
// Builds the D# descriptor per CDNA5 ISA:
//   group0: [count=1 | lds_addr | global_addr(57b) | type=2]
//   group1: [data_size=4B, pad cfg | tensor_dim0/1 | tile_dim0/1 | dim0 stride]
// pad_icode: pad after 2^(c+1) DWORDs; pad_acode: insert (c+1) DWORDs.
// Caller must be a single wave; follow with s_wait_tensorcnt + barrier.
// ---------------------------------------------------------------------------
__device__ __forceinline__ void tdm_load_2d(
    unsigned lds_off, const float* gptr,
    unsigned tensor_d0, unsigned tensor_d1, unsigned long d0_stride,
    unsigned tile_d0, unsigned tile_d1,
    unsigned pad_icode, unsigned pad_acode, bool pad_en)
{
    unsigned long ga = (unsigned long)(uintptr_t)gptr;
    u32x4 g0;
    g0[0] = 1u;                                          // count=1, user mode
    g0[1] = lds_off;                                     // lds_addr (bytes)
    g0[2] = (unsigned)(ga & 0xFFFFFFFFu);                // global_addr[31:0]
    g0[3] = (unsigned)((ga >> 32) & 0x1FFFFFFu)          // global_addr[56:32]
          | (2u << 30);                                  // type = 2 ("image")
    i32x8 g1;
    unsigned w0 = (2u << 16);                            // data_size = 4 bytes
    if (pad_en) w0 |= (1u << 20) | (pad_icode << 22) | (pad_acode << 25);
    g1[0] = (int)w0;                                     // wg_mask=0 (no cluster)
    g1[1] = (int)((tensor_d0 & 0xFFFFu) << 16);
    g1[2] = (int)(((tensor_d0 >> 16) & 0xFFFFu) | ((tensor_d1 & 0xFFFFu) << 16));
    g1[3] = (int)(((tensor_d1 >> 16) & 0xFFFFu) | (tile_d0 << 16));
    g1[4] = (int)(tile_d1 & 0xFFFFu);                    // tile_dim2 = 0
    g1[5] = (int)(unsigned)(d0_stride & 0xFFFFFFFFu);
    g1[6] = (int)((d0_stride >> 32) & 0xFFFFu);          // dim1_stride = 0 (2D)
    g1[7] = 0;
    i32x4 z4 = {0, 0, 0, 0};
#if __clang_major__ >= 23
    i32x8 z8 = {0, 0, 0, 0, 0, 0, 0, 0};
    __builtin_amdgcn_tensor_load_to_lds(g0, g1, z4, z4, z8, 0);
#else
    __builtin_amdgcn_tensor_load_to_lds(g0, g1, z4, z4, 0);
#endif
}

// ---------------------------------------------------------------------------
// Generic tiled GEMM: C[M,N] = A[M,K] @ W[K,N] + bias[N]
// M % 128 == 0, N % 128 == 0, K % 16 == 0. 256 threads = 8 waves.
// Wave grid 4x2; each wave computes a 32x64 sub-tile = 2x4 WMMA accumulators.
// A tile staged by the TDM (with HW padding -> stride 18); W tile staged
// transposed so every fragment read is a single ds_load_b64.
// ---------------------------------------------------------------------------
__global__ __launch_bounds__(256) void gemm_bias_kernel(
    const float* __restrict__ A, const float* __restrict__ W,
    const float* __restrict__ bias, float* __restrict__ C,
    int N, int K, int M)
{
    __shared__ float As[128][18];    // TDM pad: 2 DWORDs every 16
    __shared__ float BsT[128][18];   // W tile transposed: [col][k]

    const int tid  = threadIdx.x;
    const int lane = tid & 31;
    const int wave = tid >> 5;
    const int wr   = wave >> 1;      // 0..3
    const int wc   = wave & 1;       // 0..1
    const int l15  = lane & 15;
    const int hk   = (lane >> 4) << 1;   // 0 or 2
    const long blockRow = (long)blockIdx.x * 128;
    const int  blockCol = blockIdx.y * 128;
    const unsigned as_off = (unsigned)(uintptr_t)&As[0][0];

    v8f acc[2][4] = {};

    for (int k0 = 0; k0 < K; k0 += 16) {
        // Stage W tile 16x128 transposed (2 float4 loads per thread)
#pragma unroll
        for (int i = 0; i < 2; ++i) {
            int idx = tid + i * 256;
            int r = idx >> 5;                 // k row 0..15
            int c = (idx & 31) << 2;          // col 0..124
            float4 v = *(const float4*)(W + (long)(k0 + r) * N + blockCol + c);
            BsT[c + 0][r] = v.x; BsT[c + 1][r] = v.y;
            BsT[c + 2][r] = v.z; BsT[c + 3][r] = v.w;
        }
        if (k0 + 16 < K)   // prefetch next weight tile (global_prefetch_b8)
            __builtin_prefetch(W + (long)(k0 + 16 + wave) * N + blockCol + lane * 4, 0, 1);
        // TDM: A tile 128 rows x 16 cols, row stride K, padded to 18 in LDS
        if (wave == 0) {
            tdm_load_2d(as_off, A + blockRow * (long)K + k0,
                        (unsigned)K, (unsigned)M, (unsigned long)K,
                        16u, 128u, /*pad 16 dw*/3u, /*+2 dw*/1u, true);
            __builtin_amdgcn_s_wait_tensorcnt(0);
        }
        __syncthreads();

#pragma unroll
        for (int kk = 0; kk < 16; kk += 4) {
            v2f a[2], b[4];
#pragma unroll
            for (int i = 0; i < 2; ++i) {
                int row = wr * 32 + i * 16 + l15;
                a[i][0] = As[row][kk + hk];
                a[i][1] = As[row][kk + hk + 1];
            }
#pragma unroll
            for (int j = 0; j < 4; ++j) {
                int col = wc * 64 + j * 16 + l15;
                b[j][0] = BsT[col][kk + hk];
                b[j][1] = BsT[col][kk + hk + 1];
            }
#pragma unroll
            for (int i = 0; i < 2; ++i)
#pragma unroll
                for (int j = 0; j < 4; ++j)
                    acc[i][j] = __builtin_amdgcn_wmma_f32_16x16x4_f32(
                        false, a[i], false, b[j], (short)0, acc[i][j], false, false);
        }
        __syncthreads();
    }

    const int rowHalf = (lane >> 4) * 8;   // lanes 16..31 hold M = v + 8
#pragma unroll
    for (int i = 0; i < 2; ++i) {
        long rbase = blockRow + wr * 32 + i * 16 + rowHalf;
#pragma unroll
        for (int j = 0; j < 4; ++j) {
            int col = blockCol + wc * 64 + j * 16 + l15;
            float bv = bias[col];
#pragma unroll
            for (int v = 0; v < 8; ++v)
                C[(rbase + v) * (long)N + col] = acc[i][j][v] + bv;
        }
    }
}

// ---------------------------------------------------------------------------
// Per-head LayerNorm over the last HD=128 elements, in place.
// Layout [B, L, NH, HD]; one wave32 per row, 4 elements per lane.
// ---------------------------------------------------------------------------
__global__ __launch_bounds__(256) void ln_head_inplace_kernel(
    float* __restrict__ x, const float* __restrict__ g,
    const float* __restrict__ bet, long nrows)
{
    long row = (long)blockIdx.x * 8 + (threadIdx.x >> 5);
    if (row >= nrows) return;
    int lane = threadIdx.x & 31;
    float* xp = x + row * HD;

    float v0 = xp[lane], v1 = xp[lane + 32], v2 = xp[lane + 64], v3 = xp[lane + 96];
    float s = v0 + v1 + v2 + v3;
#pragma unroll
    for (int off = 16; off >= 1; off >>= 1) s += __shfl_xor(s, off, 32);
    float mean = s * (1.0f / 128.0f);
    float d0 = v0 - mean, d1 = v1 - mean, d2 = v2 - mean, d3 = v3 - mean;
    float q = d0 * d0 + d1 * d1 + d2 * d2 + d3 * d3;
#pragma unroll
    for (int off = 16; off >= 1; off >>= 1) q += __shfl_xor(q, off, 32);
    float inv = rsqrtf(q * (1.0f / 128.0f) + 1e-5f);

    xp[lane]      = d0 * inv * g[lane]      + bet[lane];
    xp[lane + 32] = d1 * inv * g[lane + 32] + bet[lane + 32];
    xp[lane + 64] = d2 * inv * g[lane + 64] + bet[lane + 64];
    xp[lane + 96] = d3 * inv * g[lane + 96] + bet[lane + 96];
}

// ---------------------------------------------------------------------------
// Scores: S[b,h,q,k] = scale * q_ln[b,q,h,:] . k_ln[b,k,h,:]
// Batched over b*NH; 64x64 output tiles, K = HD = 128 in steps of 32.
// Q and K tiles both staged by TDM (waves 0 and 1 issue parallel DMAs),
// HW padding gives the 34-float LDS row stride.
// ---------------------------------------------------------------------------
__global__ __launch_bounds__(256) void attn_scores_kernel(
    const float* __restrict__ qln,   // [B, LQ, NH, HD]
    const float* __restrict__ kln,   // [B, LK, NH, HD]
    float* __restrict__ S)           // [B*NH, LQ, LK]
{
    __shared__ float Qs[64][34];
    __shared__ float Ks[64][34];     // stored [n][k] (k_ln rows)

    const int tid  = threadIdx.x;
    const int lane = tid & 31;
    const int wave = tid >> 5;
    const int wr   = wave >> 1;      // 0..3
    const int wc   = wave & 1;       // 0..1
    const int l15  = lane & 15;
    const int hk   = (lane >> 4) << 1;
    const int m0   = blockIdx.x * 64;
    const int n0   = blockIdx.y * 64;
    const int batch = blockIdx.z;    // b*NH + h
    const int b = batch >> 4, h = batch & 15;
    const unsigned qs_off = (unsigned)(uintptr_t)&Qs[0][0];
    const unsigned ks_off = (unsigned)(uintptr_t)&Ks[0][0];

    v8f acc[2] = {};

    for (int k0 = 0; k0 < HD; k0 += 32) {
        if (wave == 0) {           // Q tile: 64 rows x 32, row stride NH*HD
            tdm_load_2d(qs_off,
                        qln + ((long)(b * LQ + m0) * NH + h) * HD + k0,
                        (unsigned)(NH * HD), (unsigned)(1u << 20),
                        (unsigned long)(NH * HD),
                        32u, 64u, /*pad 32 dw*/4u, /*+2 dw*/1u, true);
            __builtin_amdgcn_s_wait_tensorcnt(0);
        } else if (wave == 1) {    // K tile: 64 rows x 32, row stride NH*HD
            tdm_load_2d(ks_off,
                        kln + ((long)(b * LK + n0) * NH + h) * HD + k0,
                        (unsigned)(NH * HD), (unsigned)(1u << 20),
                        (unsigned long)(NH * HD),
                        32u, 64u, 4u, 1u, true);
            __builtin_amdgcn_s_wait_tensorcnt(0);
        }
        __syncthreads();

#pragma unroll
        for (int kk = 0; kk < 32; kk += 4) {
            v2f a, bb[2];
            int row = wr * 16 + l15;
            a[0] = Qs[row][kk + hk];
            a[1] = Qs[row][kk + hk + 1];
#pragma unroll
            for (int j = 0; j < 2; ++j) {
                int col = wc * 32 + j * 16 + l15;
                bb[j][0] = Ks[col][kk + hk];       // B[k][n] = k_ln[n][k]
                bb[j][1] = Ks[col][kk + hk + 1];
            }
#pragma unroll
            for (int j = 0; j < 2; ++j)
                acc[j] = __builtin_amdgcn_wmma_f32_16x16x4_f32(
                    false, a, false, bb[j], (short)0, acc[j], false, false);
        }
        __syncthreads();
    }

    const float scale = 0.08838834764831845f;   // 128^-0.5
    const int rowHalf = (lane >> 4) * 8;
    long sbase = (long)batch * LQ * LK;
#pragma unroll
    for (int j = 0; j < 2; ++j) {
        int col = n0 + wc * 32 + j * 16 + l15;
#pragma unroll
        for (int v = 0; v < 8; ++v) {
            int row = m0 + wr * 16 + rowHalf + v;
            S[sbase + (long)row * LK + col] = acc[j][v] * scale;
        }
    }
}

// ---------------------------------------------------------------------------
// Row softmax over LK = 576 (18 elements per lane), in place. 1 wave per row.
// ---------------------------------------------------------------------------
__global__ __launch_bounds__(256) void softmax_kernel(float* __restrict__ S, long nrows)
{
    long row = (long)blockIdx.x * 8 + (threadIdx.x >> 5);
    if (row >= nrows) return;
    int lane = threadIdx.x & 31;
    float* p = S + row * LK;

    float vals[18];
    float m = -3.402823466e38f;
#pragma unroll
    for (int j = 0; j < 18; ++j) { vals[j] = p[lane + j * 32]; m = fmaxf(m, vals[j]); }
#pragma unroll
    for (int off = 16; off >= 1; off >>= 1) m = fmaxf(m, __shfl_xor(m, off, 32));
    float s = 0.0f;
#pragma unroll
    for (int j = 0; j < 18; ++j) { vals[j] = expf(vals[j] - m); s += vals[j]; }
#pragma unroll
    for (int off = 16; off >= 1; off >>= 1) s += __shfl_xor(s, off, 32);
    float inv = 1.0f / s;
#pragma unroll
    for (int j = 0; j < 18; ++j) p[lane + j * 32] = vals[j] * inv;
}

// ---------------------------------------------------------------------------
// ctx[b,q,h,:] = attn[b,h,q,:] @ v[b,:,h,:]
// Batched over b*NH; 64x64 tiles, K = LK = 576 in steps of 16.
// attn tile staged by TDM (padded to stride 18); V tile staged transposed
// so fragment reads are single ds_load_b64.
// ---------------------------------------------------------------------------
__global__ __launch_bounds__(256) void attn_ctx_kernel(
    const float* __restrict__ P,    // attn [B*NH, LQ, LK]
    const float* __restrict__ V,    // v_proj [B, LK, NH, HD]
    float* __restrict__ ctx)        // [B, LQ, NH, HD]
{
    __shared__ float As[64][18];
    __shared__ float BsT[64][18];   // V tile transposed: [hd_col][k]

    const int tid  = threadIdx.x;
    const int lane = tid & 31;
    const int wave = tid >> 5;
    const int wr   = wave >> 1;
    const int wc   = wave & 1;
    const int l15  = lane & 15;
    const int hk   = (lane >> 4) << 1;
    const int m0   = blockIdx.x * 64;
    const int n0   = blockIdx.y * 64;
    const int batch = blockIdx.z;
    const int b = batch >> 4, h = batch & 15;
    const long pbase = (long)batch * LQ * LK;
    const unsigned as_off = (unsigned)(uintptr_t)&As[0][0];

    v8f acc[2] = {};

    for (int k0 = 0; k0 < LK; k0 += 16) {
        {   // V tile 16x64 transposed (1 float4 load per thread)
            int r = tid >> 4;                 // k row 0..15
            int c = (tid & 15) << 2;          // hd col 0..60
            float4 v = *(const float4*)(V + ((long)(b * LK + k0 + r) * NH + h) * HD + n0 + c);
            BsT[c + 0][r] = v.x; BsT[c + 1][r] = v.y;
            BsT[c + 2][r] = v.z; BsT[c + 3][r] = v.w;
        }
        if (wave == 0) {   // attn tile: 64 rows x 16, row stride LK, pad->18
            tdm_load_2d(as_off, P + pbase + (long)m0 * LK + k0,
                        (unsigned)LK, (unsigned)LQ, (unsigned long)LK,
                        16u, 64u, 3u, 1u, true);
            __builtin_amdgcn_s_wait_tensorcnt(0);
        }
        __syncthreads();

#pragma unroll
        for (int kk = 0; kk < 16; kk += 4) {
            v2f a, bb[2];
            int row = wr * 16 + l15;
            a[0] = As[row][kk + hk];
            a[1] = As[row][kk + hk + 1];
#pragma unroll
            for (int j = 0; j < 2; ++j) {
                int col = wc * 32 + j * 16 + l15;
                bb[j][0] = BsT[col][kk + hk];
                bb[j][1] = BsT[col][kk + hk + 1];
            }
#pragma unroll
            for (int j = 0; j < 2; ++j)
                acc[j] = __builtin_amdgcn_wmma_f32_16x16x4_f32(
                    false, a, false, bb[j], (short)0, acc[j], false, false);
        }
        __syncthreads();
    }

    const int rowHalf = (lane >> 4) * 8;
#pragma unroll
    for (int j = 0; j < 2; ++j) {
        int col = n0 + wc * 32 + j * 16 + l15;
#pragma unroll
        for (int v = 0; v < 8; ++v) {
            int row = m0 + wr * 16 + rowHalf + v;
            ctx[((long)(b * LQ + row) * NH + h) * HD + col] = acc[j][v];
        }
    }
}

// ---------------------------------------------------------------------------
extern "C" void kernel_launch(void* const* d_in, const int* in_sizes, int n_in,
                              void* d_out, int out_size, void* d_ws, size_t ws_size,
                              hipStream_t stream)
{
    const float* query = (const float*)d_in[0];
    const float* key   = (const float*)d_in[1];
    const float* Wq    = (const float*)d_in[2];
    const float* bq    = (const float*)d_in[3];
    const float* Wk    = (const float*)d_in[4];
    const float* bk    = (const float*)d_in[5];
    const float* Wv    = (const float*)d_in[6];
    const float* bv    = (const float*)d_in[7];
    const float* Wo    = (const float*)d_in[8];
    const float* bo    = (const float*)d_in[9];
    const float* gq    = (const float*)d_in[10];
    const float* betaq = (const float*)d_in[11];
    const float* gk    = (const float*)d_in[12];
    const float* betak = (const float*)d_in[13];

    float* out  = (float*)d_out;                           // [B, LQ, QD]
    float* attn = (float*)d_out + (long)B_ * LQ * QD;      // [B, NH, LQ, LK]

    float* ws  = (float*)d_ws;                             // ~210 MB used
    float* qp  = ws;                                       // [B, LQ, NH, HD]
    float* kp  = qp + (long)B_ * LQ * H_;                  // [B, LK, NH, HD]
    float* vp  = kp + (long)B_ * LK * H_;                  // [B, LK, NH, HD]
    float* ctx = vp + (long)B_ * LK * H_;                  // [B, LQ, NH, HD]

    dim3 blk(256);

    // Q/K/V projections (bias fused)
    gemm_bias_kernel<<<dim3(B_ * LQ / 128, H_ / 128), blk, 0, stream>>>(query, Wq, bq, qp, H_, QD, B_ * LQ);
    gemm_bias_kernel<<<dim3(B_ * LK / 128, H_ / 128), blk, 0, stream>>>(key,   Wk, bk, kp, H_, KD, B_ * LK);
    gemm_bias_kernel<<<dim3(B_ * LK / 128, H_ / 128), blk, 0, stream>>>(key,   Wv, bv, vp, H_, KD, B_ * LK);

    // Per-head QK LayerNorm, in place
    ln_head_inplace_kernel<<<(B_ * LQ * NH) / 8, blk, 0, stream>>>(qp, gq, betaq, (long)B_ * LQ * NH);
    ln_head_inplace_kernel<<<(B_ * LK * NH) / 8, blk, 0, stream>>>(kp, gk, betak, (long)B_ * LK * NH);

    // Scores (scale fused) written straight into the attn output region
    attn_scores_kernel<<<dim3(LQ / 64, LK / 64, B_ * NH), blk, 0, stream>>>(qp, kp, attn);

    // Softmax in place on attn
    softmax_kernel<<<(B_ * NH * LQ) / 8, blk, 0, stream>>>(attn, (long)B_ * NH * LQ);

    // attn @ V -> ctx [B, LQ, H]
    attn_ctx_kernel<<<dim3(LQ / 64, HD / 64, B_ * NH), blk, 0, stream>>>(attn, vp, ctx);

    // Output projection
    gemm_bias_kernel<<<dim3(B_ * LQ / 128, QD / 128), blk, 0, stream>>>(ctx, Wo, bo, out, QD, H_, B_ * LQ);
}